// MultiHeadAttentionBlock_23502061043960
// MI455X (gfx1250) — compile-verified
//
#include <hip/hip_runtime.h>

// MI455X (gfx1250) fused MHA block (no-softmax variant => fully linear).
// Re-bracketed:  out = q @ W4_b + (b_q @ P_b + b_o)
//   G_b  = k_b^T v_b                       [512,512] per batch
//   M_bh = (1/8)(w_k_h G_b w_v_h^T + bias-correction terms)   [64,64]
//   P_b  = blockdiag(M_b) . w_o^T          [512,512]
//   W4_b = w_q^T P_b                       [512,512]
// All GEMMs run on V_WMMA_F32_16X16X4_F32 (native fp32 matrix path, wave32).
// Dominant kernels (kG, kOut) use 32x64 register tiles: 8 WMMAs share each
// B-row load => ~1.25-1.5 vmem per wmma.

typedef float v2f __attribute__((ext_vector_type(2)));
typedef float v8f __attribute__((ext_vector_type(8)));

#define DM   512
#define SEQ  4096
#define NB   2
#define NH   8
#define DKH  64

__device__ __forceinline__ v8f wmma4(v2f a, v2f b, v8f c) {
  return __builtin_amdgcn_wmma_f32_16x16x4_f32(
      /*neg_a=*/false, a, /*neg_b=*/false, b,
      /*c_mod=*/(short)0, c, /*reuse_a=*/false, /*reuse_b=*/false);
}

// ---------- G_b[d,e] = sum_s k[b,s,d] * v[b,s,e] ------------------------
// 256 wave-jobs: each wave computes a 32(d) x 64(e) tile, K-loop over SEQ.
__global__ void __launch_bounds__(256) kG(const float* __restrict__ k,
                                          const float* __restrict__ v,
                                          float* __restrict__ G) {
  const int lane = threadIdx.x & 31, wv = threadIdx.x >> 5;
  const int lm = lane & 15, lh = lane >> 4;
  const int job = blockIdx.x * 8 + wv;        // 0..255
  const int b = job >> 7;
  const int rest = job & 127;
  const int d0 = (rest >> 3) * 32;            // 16 d-tiles of 32
  const int e0 = (rest & 7) * 64;             // 8 e-groups
  const float* kb = k + (size_t)b * SEQ * DM;
  const float* vb = v + (size_t)b * SEQ * DM;
  v8f acc[8] = {};
  #pragma unroll 2
  for (int s0 = 0; s0 < SEQ; s0 += 4) {
    size_t rb = (size_t)(s0 + 2 * lh) * DM;   // K rows strided by DM
    v2f a0; a0.x = kb[rb + d0 + lm];      a0.y = kb[rb + DM + d0 + lm];
    v2f a1; a1.x = kb[rb + d0 + 16 + lm]; a1.y = kb[rb + DM + d0 + 16 + lm];
    #pragma unroll
    for (int t = 0; t < 4; ++t) {
      v2f bm; bm.x = vb[rb + e0 + 16 * t + lm]; bm.y = vb[rb + DM + e0 + 16 * t + lm];
      acc[t]     = wmma4(a0, bm, acc[t]);
      acc[4 + t] = wmma4(a1, bm, acc[4 + t]);
    }
  }
  float* Gb = G + ((size_t)b * DM + d0) * DM + e0;
  #pragma unroll
  for (int m = 0; m < 2; ++m)
    #pragma unroll
    for (int t = 0; t < 4; ++t)
      #pragma unroll
      for (int r = 0; r < 8; ++r)
        Gb[(size_t)(16 * m + r + 8 * lh) * DM + 16 * t + lm] = acc[4 * m + t][r];
}

// ---------- T_bh[i,e] = sum_d w_k[h*64+i, d] * G_b[d,e] -----------------
__global__ void __launch_bounds__(256) kT(const float* __restrict__ wk,
                                          const float* __restrict__ G,
                                          float* __restrict__ T) {
  const int lane = threadIdx.x & 31, wv = threadIdx.x >> 5;
  const int lm = lane & 15, lh = lane >> 4;
  const int job = blockIdx.x * 8 + wv;        // 0..511
  const int bh = job >> 5;
  const int b = bh >> 3, h = bh & 7;
  const int rest = job & 31;
  const int i0 = (rest >> 3) * 16;
  const int e0 = (rest & 7) * 64;
  const float* arow = wk + (size_t)(h * 64 + i0 + lm) * DM + 2 * lh;  // contiguous K
  const float* Gb = G + (size_t)b * DM * DM;
  v8f acc[4] = {};
  for (int kk = 0; kk < DM; kk += 4) {
    v2f a = *(const v2f*)(arow + kk);
    size_t rb = (size_t)(kk + 2 * lh) * DM;
    #pragma unroll
    for (int t = 0; t < 4; ++t) {
      v2f bm; bm.x = Gb[rb + e0 + 16 * t + lm]; bm.y = Gb[rb + DM + e0 + 16 * t + lm];
      acc[t] = wmma4(a, bm, acc[t]);
    }
  }
  float* Tb = T + ((size_t)bh * 64 + i0) * DM + e0;
  #pragma unroll
  for (int t = 0; t < 4; ++t)
    #pragma unroll
    for (int r = 0; r < 8; ++r)
      Tb[(size_t)(r + 8 * lh) * DM + 16 * t + lm] = acc[t][r];
}

// ---------- M_bh[i,j] = (1/8)(sum_e T_bh[i,e] w_v[h*64+j,e] + bias terms)
__global__ void __launch_bounds__(256) kM(const float* __restrict__ T,
                                          const float* __restrict__ wvw,
                                          const float* __restrict__ bk,
                                          const float* __restrict__ bv,
                                          const float* __restrict__ u,   // [B,512] = w_k . ksum
                                          const float* __restrict__ z,   // [B,512] = w_v . vsum
                                          float* __restrict__ M) {
  const int lane = threadIdx.x & 31, wv = threadIdx.x >> 5;
  const int lm = lane & 15, lh = lane >> 4;
  const int job = blockIdx.x * 8 + wv;        // 0..63
  const int bh = job >> 2;
  const int b = bh >> 3, h = bh & 7;
  const int i0 = (job & 3) * 16;
  const float* arow = T + ((size_t)bh * 64 + i0 + lm) * DM + 2 * lh;
  v8f acc[4] = {};
  for (int kk = 0; kk < DM; kk += 4) {
    v2f a = *(const v2f*)(arow + kk);
    #pragma unroll
    for (int t = 0; t < 4; ++t) {
      v2f bm = *(const v2f*)(wvw + (size_t)(h * 64 + 16 * t + lm) * DM + kk + 2 * lh);
      acc[t] = wmma4(a, bm, acc[t]);
    }
  }
  #pragma unroll
  for (int t = 0; t < 4; ++t) {
    const int jf = h * 64 + 16 * t + lm;
    const float bvj = bv[jf];
    const float zj  = z[(size_t)b * DM + jf];
    #pragma unroll
    for (int r = 0; r < 8; ++r) {
      const int ifl = h * 64 + i0 + r + 8 * lh;
      const float bki = bk[ifl];
      const float ui  = u[(size_t)b * DM + ifl];
      float val = acc[t][r] + ui * bvj + bki * zj + (float)SEQ * bki * bvj;
      M[((size_t)bh * 64 + i0 + r + 8 * lh) * DKH + 16 * t + lm] = 0.125f * val;
    }
  }
}

// ---------- P_b[h*64+i, o] = sum_j M_bh[i,j] * w_o[o, h*64+j] -----------
__global__ void __launch_bounds__(256) kP(const float* __restrict__ M,
                                          const float* __restrict__ wo,
                                          float* __restrict__ P) {
  const int lane = threadIdx.x & 31, wv = threadIdx.x >> 5;
  const int lm = lane & 15, lh = lane >> 4;
  const int job = blockIdx.x * 8 + wv;        // 0..511
  const int bh = job >> 5;
  const int b = bh >> 3, h = bh & 7;
  const int i0 = ((job >> 3) & 3) * 16;
  const int o0 = (job & 7) * 64;
  const float* arow = M + ((size_t)bh * 64 + i0 + lm) * DKH + 2 * lh;
  v8f acc[4] = {};
  for (int kk = 0; kk < DKH; kk += 4) {
    v2f a = *(const v2f*)(arow + kk);
    #pragma unroll
    for (int t = 0; t < 4; ++t) {
      v2f bm = *(const v2f*)(wo + (size_t)(o0 + 16 * t + lm) * DM + h * 64 + kk + 2 * lh);
      acc[t] = wmma4(a, bm, acc[t]);
    }
  }
  float* Pb = P + ((size_t)b * DM + h * 64 + i0) * DM + o0;
  #pragma unroll
  for (int t = 0; t < 4; ++t)
    #pragma unroll
    for (int r = 0; r < 8; ++r)
      Pb[(size_t)(r + 8 * lh) * DM + 16 * t + lm] = acc[t][r];
}

// ---------- W4_b[d,o] = sum_f w_q[f,d] * P_b[f,o] -----------------------
__global__ void __launch_bounds__(256) kW4(const float* __restrict__ wq,
                                           const float* __restrict__ P,
                                           float* __restrict__ W4) {
  const int lane = threadIdx.x & 31, wv = threadIdx.x >> 5;
  const int lm = lane & 15, lh = lane >> 4;
  const int job = blockIdx.x * 8 + wv;        // 0..511
  const int b = job >> 8;
  const int rest = job & 255;
  const int d0 = (rest >> 3) * 16;
  const int o0 = (rest & 7) * 64;
  const float* Pb = P + (size_t)b * DM * DM;
  v8f acc[4] = {};
  for (int kk = 0; kk < DM; kk += 4) {
    size_t fr = (size_t)(kk + 2 * lh) * DM;
    v2f a; a.x = wq[fr + d0 + lm]; a.y = wq[fr + DM + d0 + lm];
    #pragma unroll
    for (int t = 0; t < 4; ++t) {
      v2f bm; bm.x = Pb[fr + o0 + 16 * t + lm]; bm.y = Pb[fr + DM + o0 + 16 * t + lm];
      acc[t] = wmma4(a, bm, acc[t]);
    }
  }
  float* Wb = W4 + ((size_t)b * DM + d0) * DM + o0;
  #pragma unroll
  for (int t = 0; t < 4; ++t)
    #pragma unroll
    for (int r = 0; r < 8; ++r)
      Wb[(size_t)(r + 8 * lh) * DM + 16 * t + lm] = acc[t][r];
}

// ---------- out[b,s,o] = sum_d q[b,s,d] W4_b[d,o] + cq[b,o] + b_o[o] ----
// 256 wave-jobs per o-group: each wave computes a 32(s) x 64(o) tile.
__global__ void __launch_bounds__(256) kOut(const float* __restrict__ q,
                                            const float* __restrict__ W4,
                                            const float* __restrict__ bo,
                                            const float* __restrict__ cq,
                                            float* __restrict__ out) {
  const int lane = threadIdx.x & 31, wv = threadIdx.x >> 5;
  const int lm = lane & 15, lh = lane >> 4;
  const int job = blockIdx.x * 8 + wv;        // 0..255 m-tiles of 32 rows
  const int b = job >> 7;
  const int n0 = (job & 127) * 32;
  const int o0 = blockIdx.y * 64;
  const float* qrow0 = q + ((size_t)b * SEQ + n0 + lm) * DM + 2 * lh;
  const float* qrow1 = qrow0 + (size_t)16 * DM;
  const float* Wb = W4 + (size_t)b * DM * DM;
  v8f acc[8] = {};
  #pragma unroll 2
  for (int kk = 0; kk < DM; kk += 4) {
    v2f a0 = *(const v2f*)(qrow0 + kk);       // contiguous K => float2
    v2f a1 = *(const v2f*)(qrow1 + kk);
    size_t rb = (size_t)(kk + 2 * lh) * DM;
    #pragma unroll
    for (int t = 0; t < 4; ++t) {
      v2f bm; bm.x = Wb[rb + o0 + 16 * t + lm]; bm.y = Wb[rb + DM + o0 + 16 * t + lm];
      acc[t]     = wmma4(a0, bm, acc[t]);
      acc[4 + t] = wmma4(a1, bm, acc[4 + t]);
    }
  }
  float* ob = out + ((size_t)b * SEQ + n0) * DM + o0;
  #pragma unroll
  for (int t = 0; t < 4; ++t) {
    float add = bo[o0 + 16 * t + lm] + cq[(size_t)b * DM + o0 + 16 * t + lm];
    #pragma unroll
    for (int m = 0; m < 2; ++m)
      #pragma unroll
      for (int r = 0; r < 8; ++r)
        ob[(size_t)(16 * m + r + 8 * lh) * DM + 16 * t + lm] = acc[4 * m + t][r] + add;
  }
}

// ---------- small helper kernels (bias-correction precomputes) ----------
__global__ void kSums(const float* __restrict__ k, const float* __restrict__ v,
                      float* __restrict__ ksum, float* __restrict__ vsum) {
  int idx = blockIdx.x * blockDim.x + threadIdx.x;  // [B*512]
  if (idx >= NB * DM) return;
  int b = idx >> 9, d = idx & (DM - 1);
  const float* kp = k + (size_t)b * SEQ * DM + d;
  const float* vp = v + (size_t)b * SEQ * DM + d;
  float sk = 0.f, sv = 0.f;
  for (int s = 0; s < SEQ; ++s) { sk += kp[(size_t)s * DM]; sv += vp[(size_t)s * DM]; }
  ksum[idx] = sk; vsum[idx] = sv;
}

__global__ void kUZ(const float* __restrict__ wk, const float* __restrict__ wvw,
                    const float* __restrict__ ksum, const float* __restrict__ vsum,
                    float* __restrict__ u, float* __restrict__ z) {
  int idx = blockIdx.x * blockDim.x + threadIdx.x;  // [B*512]
  if (idx >= NB * DM) return;
  int b = idx >> 9, f = idx & (DM - 1);
  const float* wkr = wk + (size_t)f * DM;
  const float* wvr = wvw + (size_t)f * DM;
  const float* ks = ksum + (size_t)b * DM;
  const float* vs = vsum + (size_t)b * DM;
  float su = 0.f, sz = 0.f;
  for (int d = 0; d < DM; ++d) { su += wkr[d] * ks[d]; sz += wvr[d] * vs[d]; }
  u[idx] = su; z[idx] = sz;
}

__global__ void kCQ(const float* __restrict__ bq, const float* __restrict__ P,
                    float* __restrict__ cq) {
  int idx = blockIdx.x * blockDim.x + threadIdx.x;  // [B*512]
  if (idx >= NB * DM) return;
  int b = idx >> 9, o = idx & (DM - 1);
  const float* Pb = P + (size_t)b * DM * DM + o;
  float s = 0.f;
  for (int f = 0; f < DM; ++f) s += bq[f] * Pb[(size_t)f * DM];
  cq[idx] = s;
}

extern "C" void kernel_launch(void* const* d_in, const int* in_sizes, int n_in,
                              void* d_out, int out_size, void* d_ws, size_t ws_size,
                              hipStream_t stream) {
  const float* q   = (const float*)d_in[0];
  const float* k   = (const float*)d_in[1];
  const float* v   = (const float*)d_in[2];
  const float* wq  = (const float*)d_in[3];
  const float* bq  = (const float*)d_in[4];
  const float* wk  = (const float*)d_in[5];
  const float* bk  = (const float*)d_in[6];
  const float* wvw = (const float*)d_in[7];
  const float* bv  = (const float*)d_in[8];
  const float* wo  = (const float*)d_in[9];
  const float* bo  = (const float*)d_in[10];
  float* out = (float*)d_out;

  float* ws = (float*)d_ws;          // ~8.7 MB total
  float* G    = ws;                  // [B,512,512]
  float* T    = G    + (size_t)NB * DM * DM;      // [B*H,64,512]
  float* M    = T    + (size_t)NB * DM * DM;      // [B*H,64,64]
  float* P    = M    + (size_t)NB * NH * DKH * DKH;
  float* W4   = P    + (size_t)NB * DM * DM;
  float* ksum = W4   + (size_t)NB * DM * DM;
  float* vsum = ksum + (size_t)NB * DM;
  float* u    = vsum + (size_t)NB * DM;
  float* z    = u    + (size_t)NB * DM;
  float* cq   = z    + (size_t)NB * DM;

  kSums<<<4, 256, 0, stream>>>(k, v, ksum, vsum);
  kUZ  <<<4, 256, 0, stream>>>(wk, wvw, ksum, vsum, u, z);
  kG   <<<32, 256, 0, stream>>>(k, v, G);
  kT   <<<64, 256, 0, stream>>>(wk, G, T);
  kM   <<<8, 256, 0, stream>>>(T, wvw, bk, bv, u, z, M);
  kP   <<<64, 256, 0, stream>>>(M, wo, P);
  kCQ  <<<4, 256, 0, stream>>>(bq, P, cq);
  kW4  <<<64, 256, 0, stream>>>(wq, P, W4);
  dim3 go(32, 8, 1);
  kOut <<<go, 256, 0, stream>>>(q, W4, bo, cq, out);
}